// FusionLayer_70892730188024
// MI455X (gfx1250) — compile-verified
//
#include <hip/hip_runtime.h>
#include <math.h>

#define NNODES 50000
#define NEDGES 800000
#define DIM    128
#define NTOK   9
#define LN_EPS 1e-5f

typedef __attribute__((ext_vector_type(2))) float v2f;
typedef __attribute__((ext_vector_type(8))) float v8f;

// ---------------- float atomic max/min via sign-split trick ----------------
__device__ __forceinline__ void atomicMaxF(float* addr, float v) {
  if (v >= 0.0f) atomicMax((int*)addr, __float_as_int(v));
  else           atomicMin((unsigned int*)addr, __float_as_uint(v));
}
__device__ __forceinline__ void atomicMinF(float* addr, float v) {
  if (v >= 0.0f) atomicMin((int*)addr, __float_as_int(v));
  else           atomicMax((unsigned int*)addr, __float_as_uint(v));
}

// ---------------- 1) init tokens (self token + neutral agg) ----------------
__global__ void init_kernel(const float* __restrict__ x, float* __restrict__ tok,
                            float* __restrict__ cnt0, float* __restrict__ cnt1) {
  int idx  = blockIdx.x * blockDim.x + threadIdx.x;
  int node = idx >> 5;
  int d    = (idx & 31) << 2;
  float4 xs = *(const float4*)(x + (size_t)node * DIM + d);
  float ninf = __int_as_float(0xff800000);
  float pinf = __int_as_float(0x7f800000);
  float4 vni = make_float4(ninf, ninf, ninf, ninf);
  float4 vpi = make_float4(pinf, pinf, pinf, pinf);
  float4 vz  = make_float4(0.f, 0.f, 0.f, 0.f);
  float* base = tok + (size_t)node * NTOK * DIM + d;
  *(float4*)(base + 0 * DIM) = xs;   // self
  *(float4*)(base + 1 * DIM) = vni;  // max0
  *(float4*)(base + 2 * DIM) = vpi;  // min0
  *(float4*)(base + 3 * DIM) = vz;   // sum0
  *(float4*)(base + 4 * DIM) = vz;   // mean0
  *(float4*)(base + 5 * DIM) = vni;  // max1
  *(float4*)(base + 6 * DIM) = vpi;  // min1
  *(float4*)(base + 7 * DIM) = vz;   // sum1
  *(float4*)(base + 8 * DIM) = vz;   // mean1
  if ((idx & 31) == 0) { cnt0[node] = 0.f; cnt1[node] = 0.f; }
}

// ---------------- 2) edge scatter: one wave32 per edge ----------------
__global__ void edge_kernel(const float* __restrict__ x, const int* __restrict__ ei,
                            float* __restrict__ tok, float* __restrict__ cnt, int sbase) {
  int t    = blockIdx.x * blockDim.x + threadIdx.x;
  int e    = t >> 5;
  int lane = t & 31;
  int d    = lane << 2;
  int src  = ei[e];
  int dst  = ei[NEDGES + e];
  float4 m = *(const float4*)(x + (size_t)src * DIM + d);
  float* base = tok + ((size_t)dst * NTOK + sbase) * DIM + d;
  float* pmax = base;
  float* pmin = base + DIM;
  float* psum = base + 2 * DIM;
  atomicMaxF(pmax + 0, m.x); atomicMaxF(pmax + 1, m.y);
  atomicMaxF(pmax + 2, m.z); atomicMaxF(pmax + 3, m.w);
  atomicMinF(pmin + 0, m.x); atomicMinF(pmin + 1, m.y);
  atomicMinF(pmin + 2, m.z); atomicMinF(pmin + 3, m.w);
  atomicAdd(psum + 0, m.x);  atomicAdd(psum + 1, m.y);
  atomicAdd(psum + 2, m.z);  atomicAdd(psum + 3, m.w);
  if (lane == 0) atomicAdd(cnt + dst, 1.0f);
}

// ---------------- 3) finalize: mean, zero-out isolated nodes ----------------
__global__ void finalize_kernel(float* __restrict__ tok, const float* __restrict__ cnt0,
                                const float* __restrict__ cnt1) {
  int idx  = blockIdx.x * blockDim.x + threadIdx.x;
  int node = idx >> 5;
  int d    = (idx & 31) << 2;
  float* base = tok + (size_t)node * NTOK * DIM + d;
  float4 vz = make_float4(0.f, 0.f, 0.f, 0.f);
  float c0 = cnt0[node];
  if (c0 > 0.f) {
    float4 s = *(float4*)(base + 3 * DIM);
    float inv = 1.0f / c0;
    *(float4*)(base + 4 * DIM) = make_float4(s.x * inv, s.y * inv, s.z * inv, s.w * inv);
  } else {
    *(float4*)(base + 1 * DIM) = vz; *(float4*)(base + 2 * DIM) = vz;
    *(float4*)(base + 3 * DIM) = vz; *(float4*)(base + 4 * DIM) = vz;
  }
  float c1 = cnt1[node];
  if (c1 > 0.f) {
    float4 s = *(float4*)(base + 7 * DIM);
    float inv = 1.0f / c1;
    *(float4*)(base + 8 * DIM) = make_float4(s.x * inv, s.y * inv, s.z * inv, s.w * inv);
  } else {
    *(float4*)(base + 5 * DIM) = vz; *(float4*)(base + 6 * DIM) = vz;
    *(float4*)(base + 7 * DIM) = vz; *(float4*)(base + 8 * DIM) = vz;
  }
}

// ---------------- 4) GEMM: Y[rows,128] = X[rows,128] @ W^T + b ----------------
// Block = 64 rows (4 row-subtiles per wave reusing one B register pair),
// 8 waves = 8 column tiles. A tile staged via GLOBAL_LOAD_ASYNC_TO_LDS_B128
// (ASYNCcnt path), compute on V_WMMA_F32_16X16X4_F32 (exact fp32).
__launch_bounds__(256)
__global__ void gemm128_kernel(const float* __restrict__ X, const float* __restrict__ W,
                               const float* __restrict__ bias, float* __restrict__ Y,
                               int nrows) {
  __shared__ float atile[64][132];  // +4 pad: bank-conflict-free ds loads
  int tid = threadIdx.x;
  long row0 = (long)blockIdx.x * 64;
  if (row0 + 64 > nrows) row0 = (long)nrows - 64;  // overlap writes identical values

  // ---- async global -> LDS copy of the 64x128 A tile (no VGPR staging) ----
  {
    int r = tid >> 2;               // 0..63
    int c = (tid & 3) << 5;         // 0,32,64,96
    const float* gp = X + (row0 + r) * DIM + c;
    unsigned la = (unsigned)(size_t)(void*)&atile[r][c];
#pragma unroll
    for (int j = 0; j < 8; ++j) {
      unsigned long long ga = (unsigned long long)(const void*)(gp + (j << 2));
      unsigned ldsa = la + (unsigned)(j << 4);
      asm volatile("global_load_async_to_lds_b128 %0, %1, off"
                   :: "v"(ldsa), "v"(ga) : "memory");
    }
  }
  asm volatile("s_wait_asynccnt 0" ::: "memory");
  __syncthreads();

  int lane = tid & 31;
  int wv   = tid >> 5;              // column tile 0..7
  int m    = lane & 15;
  int kb   = (lane >> 4) << 1;      // 0 or 2 (f32 A/B per-lane K offset)
  int n    = (wv << 4) | m;         // output column 0..127

  v8f acc[4];
  float bv = bias[n];
#pragma unroll
  for (int rt = 0; rt < 4; ++rt)
#pragma unroll
    for (int r = 0; r < 8; ++r) acc[rt][r] = bv;

  const float* wrow = W + (size_t)n * DIM;
#pragma unroll
  for (int k0 = 0; k0 < DIM; k0 += 4) {
    float2 wb = *(const float2*)(wrow + k0 + kb);  // B[k][n] = W[n][k]
    v2f b; b[0] = wb.x; b[1] = wb.y;
#pragma unroll
    for (int rt = 0; rt < 4; ++rt) {
      v2f a;
      a[0] = atile[(rt << 4) + m][k0 + kb];
      a[1] = atile[(rt << 4) + m][k0 + kb + 1];
      acc[rt] = __builtin_amdgcn_wmma_f32_16x16x4_f32(
          false, a, false, b, (short)0, acc[rt], false, false);
    }
  }

  int mo = (lane >> 4) << 3;        // D rows: r (lanes 0-15), r+8 (lanes 16-31)
#pragma unroll
  for (int rt = 0; rt < 4; ++rt)
#pragma unroll
    for (int r = 0; r < 8; ++r)
      Y[(row0 + (rt << 4) + r + mo) * DIM + n] = acc[rt][r];
}

// ---------------- 5) attention over S=9 tokens, wave32 per node ----------------
__launch_bounds__(256)
__global__ void attn_kernel(const float* __restrict__ q0, const float* __restrict__ K,
                            const float* __restrict__ V, const float* __restrict__ cnt0,
                            const float* __restrict__ cnt1, float* __restrict__ o0) {
  int tid  = threadIdx.x;
  int lane = tid & 31;
  int node = blockIdx.x * 8 + (tid >> 5);
  int d    = lane << 2;             // lane owns 4 contiguous dims; head h = lane/8
  float4 q = *(const float4*)(q0 + (size_t)node * DIM + d);
  bool has0 = cnt0[node] > 0.f;
  bool has1 = cnt1[node] > 0.f;
  const float scale = 0.1767766952966369f;  // 1/sqrt(32)
  size_t base = (size_t)node * NTOK * DIM + d;

  float sc[NTOK];
#pragma unroll
  for (int t = 0; t < NTOK; ++t) {
    float4 kk = *(const float4*)(K + base + (size_t)t * DIM);
    float p = q.x * kk.x + q.y * kk.y + q.z * kk.z + q.w * kk.w;
    p += __shfl_xor(p, 1, 8);       // reduce within 8-lane head group (wave32)
    p += __shfl_xor(p, 2, 8);
    p += __shfl_xor(p, 4, 8);
    bool masked = (t >= 1 && t <= 4) ? !has0 : (t >= 5 ? !has1 : false);
    sc[t] = masked ? -1e30f : p * scale;
  }
  float mx = sc[0];
#pragma unroll
  for (int t = 1; t < NTOK; ++t) mx = fmaxf(mx, sc[t]);
  float ssum = 0.f;
#pragma unroll
  for (int t = 0; t < NTOK; ++t) { sc[t] = expf(sc[t] - mx); ssum += sc[t]; }
  float inv = 1.0f / ssum;

  float4 acc = make_float4(0.f, 0.f, 0.f, 0.f);
#pragma unroll
  for (int t = 0; t < NTOK; ++t) {
    float4 vv = *(const float4*)(V + base + (size_t)t * DIM);
    float p = sc[t] * inv;
    acc.x += p * vv.x; acc.y += p * vv.y; acc.z += p * vv.z; acc.w += p * vv.w;
  }
  *(float4*)(o0 + (size_t)node * DIM + d) = acc;
}

// ---------------- 6) LayerNorm, wave32 per node ----------------
__launch_bounds__(256)
__global__ void ln_kernel(const float* __restrict__ Xp, const float* __restrict__ gamma,
                          const float* __restrict__ beta, float* __restrict__ out) {
  int tid  = threadIdx.x;
  int lane = tid & 31;
  int node = blockIdx.x * 8 + (tid >> 5);
  int d    = lane << 2;
  float4 v = *(const float4*)(Xp + (size_t)node * DIM + d);
  float s  = v.x + v.y + v.z + v.w;
  float s2 = v.x * v.x + v.y * v.y + v.z * v.z + v.w * v.w;
#pragma unroll
  for (int m = 1; m < 32; m <<= 1) {
    s  += __shfl_xor(s, m, 32);
    s2 += __shfl_xor(s2, m, 32);
  }
  float mean = s * (1.0f / 128.0f);
  float var  = s2 * (1.0f / 128.0f) - mean * mean;
  float rstd = rsqrtf(var + LN_EPS);
  float4 g = *(const float4*)(gamma + d);
  float4 b = *(const float4*)(beta + d);
  float4 o;
  o.x = (v.x - mean) * rstd * g.x + b.x;
  o.y = (v.y - mean) * rstd * g.y + b.y;
  o.z = (v.z - mean) * rstd * g.z + b.z;
  o.w = (v.w - mean) * rstd * g.w + b.w;
  *(float4*)(out + (size_t)node * DIM + d) = o;
}

// ---------------- driver ----------------
extern "C" void kernel_launch(void* const* d_in, const int* in_sizes, int n_in,
                              void* d_out, int out_size, void* d_ws, size_t ws_size,
                              hipStream_t stream) {
  const float* x     = (const float*)d_in[0];
  const int*   e0    = (const int*)d_in[1];
  const int*   e1    = (const int*)d_in[2];
  const float* w_in  = (const float*)d_in[3];  // [384,128]: qw,kw,vw
  const float* b_in  = (const float*)d_in[4];  // [384]
  const float* w_out = (const float*)d_in[5];  // [128,128]
  const float* b_out = (const float*)d_in[6];
  const float* g     = (const float*)d_in[7];
  const float* be    = (const float*)d_in[8];

  float* ws   = (float*)d_ws;
  float* tok  = ws;                       // [50000, 9, 128]
  float* Kb   = ws + 57600000ull;         // [450000, 128]
  float* Vb   = ws + 115200000ull;        // [450000, 128]
  float* q0   = ws + 172800000ull;        // [50000, 128]
  float* o0   = ws + 179200000ull;        // [50000, 128]
  float* cnt0 = ws + 185600000ull;        // [50000]
  float* cnt1 = ws + 185650000ull;        // [50000]
  float* pre  = q0;                       // reuse q0 for pre-LN (q dead after attn)

  init_kernel<<<6250, 256, 0, stream>>>(x, tok, cnt0, cnt1);
  edge_kernel<<<100000, 256, 0, stream>>>(x, e0, tok, cnt0, 1);
  edge_kernel<<<100000, 256, 0, stream>>>(x, e1, tok, cnt1, 5);
  finalize_kernel<<<6250, 256, 0, stream>>>(tok, cnt0, cnt1);

  // 64-row blocks: ceil-div grids; last block clamps row0 (idempotent overlap)
  gemm128_kernel<<<782, 256, 0, stream>>>(x, w_in, b_in, q0, NNODES);                 // Q(self)
  gemm128_kernel<<<7032, 256, 0, stream>>>(tok, w_in + 16384, b_in + 128, Kb,
                                           NNODES * NTOK);                            // K
  gemm128_kernel<<<7032, 256, 0, stream>>>(tok, w_in + 32768, b_in + 256, Vb,
                                           NNODES * NTOK);                            // V

  attn_kernel<<<6250, 256, 0, stream>>>(q0, Kb, Vb, cnt0, cnt1, o0);
  gemm128_kernel<<<782, 256, 0, stream>>>(o0, w_out, b_out, pre, NNODES);             // out proj
  ln_kernel<<<6250, 256, 0, stream>>>(pre, g, be, (float*)d_out);
}